// Apply2DDispField_66838281060467
// MI455X (gfx1250) — compile-verified
//
#include <hip/hip_runtime.h>

// Batched bilinear warp (grid-sample with displacement field), MI455X / gfx1250.
// Memory-bound: ~268 MB ideal traffic -> ~11.5 us floor at 23.3 TB/s HBM.
// Image (67 MB) stays resident in the 192 MB L2 for the random gathers (RT policy);
// displacement stream + output stream use non-temporal policy so they don't evict it.
// 4 pixels per lane: 2x global_load_b128(NT) + 16x gather b32(RT) + 1x global_store_b128(NT).
// VALU trimmed (float-domain clamps, OR-based pad check, shared row offsets) so
// compute stays under the HBM floor.

typedef __attribute__((ext_vector_type(4))) float v4f;

namespace {
constexpr int kB = 16;
constexpr int kH = 1024;
constexpr int kW = 1024;
constexpr int kHW = kH * kW;

__device__ __forceinline__ float bilinear_one(const float* __restrict__ imgb,
                                              float x, float y) {
    float xf = floorf(x);
    float yf = floorf(y);

    // Clamp corners to [0, 1024] in float domain (exact; matches reference's
    // int-clamp-then-cast numerics bit-for-bit for these magnitudes).
    float x0c = fminf(fmaxf(xf,        0.0f), 1024.0f);
    float x1c = fminf(fmaxf(xf + 1.0f, 0.0f), 1024.0f);
    float y0c = fminf(fmaxf(yf,        0.0f), 1024.0f);
    float y1c = fminf(fmaxf(yf + 1.0f, 0.0f), 1024.0f);

    int x0 = (int)x0c, x1 = (int)x1c;       // exact, in [0,1024]
    int y0 = (int)y0c, y1 = (int)y1c;

    // Safe addresses: clamp to last real row/col; shared row offsets.
    int rx0 = min(x0, kH - 1) << 10;        // *1024
    int rx1 = min(x1, kH - 1) << 10;
    int cy0 = min(y0, kW - 1);
    int cy1 = min(y1, kW - 1);

    float I00 = imgb[rx0 + cy0];
    float I01 = imgb[rx0 + cy1];
    float I10 = imgb[rx1 + cy0];
    float I11 = imgb[rx1 + cy1];

    // Pad (index 1024) -> zero. Corners are >= 0, so (a|b) > 1023 <=> either OOB.
    I00 = ((x0 | y0) > 1023) ? 0.0f : I00;
    I01 = ((x0 | y1) > 1023) ? 0.0f : I01;
    I10 = ((x1 | y0) > 1023) ? 0.0f : I10;
    I11 = ((x1 | y1) > 1023) ? 0.0f : I11;

    // Weights from CLAMPED corners, as in the reference.
    float dx0 = x - x0c, dx1 = x1c - x;
    float dy0 = y - y0c, dy1 = y1c - y;

    float acc = (dx1 * dy1) * I00;
    acc = fmaf(dx1 * dy0, I01, acc);
    acc = fmaf(dx0 * dy1, I10, acc);
    acc = fmaf(dx0 * dy0, I11, acc);
    return acc;
}
} // namespace

__global__ __launch_bounds__(256)
void Apply2DDispField_warp_kernel(const float* __restrict__ img,
                                  const v4f* __restrict__ disp4,
                                  v4f* __restrict__ out4,
                                  int n_quads) {
    int gid = blockIdx.x * blockDim.x + threadIdx.x;
    if (gid >= n_quads) return;

    int p  = gid << 2;               // first pixel linear index (multiple of 4)
    int wj = p & (kW - 1);           // column of first pixel; wj..wj+3 same row
    int hi = (p >> 10) & (kH - 1);   // row
    int b  = p >> 20;                // batch

    // Streaming read of 4 displacement pairs (2x b128, NT).
    v4f dA = __builtin_nontemporal_load(&disp4[2 * gid]);      // (d0x,d0y,d1x,d1y)
    v4f dB = __builtin_nontemporal_load(&disp4[2 * gid + 1]);  // (d2x,d2y,d3x,d3y)

    const float step  = 2.0f / 1023.0f;   // linspace(-1,1,1024) step
    const float scale = 0.5f * 1023.0f;   // 0.5 * (max-1)

    float x_t = -1.0f + (float)hi * step;             // shared by all 4 pixels
    float wjf = (float)wj;
    float yt0 = -1.0f + (wjf        ) * step;
    float yt1 = -1.0f + (wjf + 1.0f) * step;
    float yt2 = -1.0f + (wjf + 2.0f) * step;
    float yt3 = -1.0f + (wjf + 3.0f) * step;

    const float* imgb = img + (size_t)b * kHW;

    // x = 0.5 * (x_t - disp + 1) * 1023  (reference formula)
    float x0p = ((x_t - dA.x) + 1.0f) * scale;
    float y0p = ((yt0 - dA.y) + 1.0f) * scale;
    float x1p = ((x_t - dA.z) + 1.0f) * scale;
    float y1p = ((yt1 - dA.w) + 1.0f) * scale;
    float x2p = ((x_t - dB.x) + 1.0f) * scale;
    float y2p = ((yt2 - dB.y) + 1.0f) * scale;
    float x3p = ((x_t - dB.z) + 1.0f) * scale;
    float y3p = ((yt3 - dB.w) + 1.0f) * scale;

    v4f r;
    r.x = bilinear_one(imgb, x0p, y0p);
    r.y = bilinear_one(imgb, x1p, y1p);
    r.z = bilinear_one(imgb, x2p, y2p);
    r.w = bilinear_one(imgb, x3p, y3p);

    // Streaming write; never re-read -> non-temporal b128.
    __builtin_nontemporal_store(r, &out4[gid]);
}

extern "C" void kernel_launch(void* const* d_in, const int* in_sizes, int n_in,
                              void* d_out, int out_size, void* d_ws, size_t ws_size,
                              hipStream_t stream) {
    (void)in_sizes; (void)n_in; (void)d_ws; (void)ws_size;

    const float* img   = (const float*)d_in[0];   // (B,H,W,1) f32
    const v4f*   disp4 = (const v4f*)d_in[1];     // (B,H,W,2) f32, 2 px per v4f
    v4f*         out4  = (v4f*)d_out;             // (B,H,W,1) f32, 4 px per v4f

    int n_quads = (kB * kHW) / 4;                 // 4,194,304 lane-tasks
    int block   = 256;                            // 8 waves (wave32) per block
    int grid    = (n_quads + block - 1) / block;

    Apply2DDispField_warp_kernel<<<grid, block, 0, stream>>>(img, disp4, out4, n_quads);
}